// GNNSafe_14602888806541
// MI455X (gfx1250) — compile-verified
//
#include <hip/hip_runtime.h>
#include <hip/hip_bf16.h>
#include <math.h>

typedef __attribute__((ext_vector_type(2))) float v2f;
typedef __attribute__((ext_vector_type(8))) float v8f;

#define C_CLS 40
#define H_DIM 64
#define ALPHA 0.5f

// ds_swizzle group-of-32 xor pattern: offset[14:10]=xor, [4:0]=and_mask=0x1f
#define SWZ_XOR(v, mask) \
    __int_as_float(__builtin_amdgcn_ds_swizzle(__float_as_int(v), (0x1f | ((mask) << 10))))

// ---------------------------------------------------------------- utilities
__global__ void k_zero(float* __restrict__ a, int n) {
    int i = blockIdx.x * blockDim.x + threadIdx.x;
    if (i < n) a[i] = 0.0f;
}

// ------------------------------------------------ pass 1: logits -> lnll, e0
// One node per thread; 40 floats = ten float4 loads (160B, 16B-aligned).
// l_nll = lse - max ; l_energy(init) = -lse   (T == 1)
__global__ void k_logits(const float* __restrict__ logits,
                         float* __restrict__ lnll,
                         float* __restrict__ ener, int n) {
    int i = blockIdx.x * blockDim.x + threadIdx.x;
    if (i >= n) return;
    const float4* p = (const float4*)(logits + (size_t)i * C_CLS);
    float4 v[10];
    float m = -INFINITY;
#pragma unroll
    for (int j = 0; j < 10; ++j) {
        v[j] = p[j];
        m = fmaxf(m, fmaxf(fmaxf(v[j].x, v[j].y), fmaxf(v[j].z, v[j].w)));
    }
    float s = 0.0f;
#pragma unroll
    for (int j = 0; j < 10; ++j) {
        s += __expf(v[j].x - m) + __expf(v[j].y - m) +
             __expf(v[j].z - m) + __expf(v[j].w - m);
    }
    float lse = m + __logf(s);
    lnll[i] = lse - m;   // -max(log_softmax)
    ener[i] = -lse;      // -logsumexp
}

// ------------------------------------------------ propagation, pass 1
// Fused: raw message accumulation AND degree count in one edge sweep.
// dinv is applied per-node in the combine (1/deg[col] is constant per target),
// so no per-edge dinv gather is needed at all.
__global__ void k_prop1(const int* __restrict__ row, const int* __restrict__ col,
                        const float* __restrict__ e,
                        float* __restrict__ msg, float* __restrict__ deg, int E) {
    int i = blockIdx.x * blockDim.x + threadIdx.x;
    if (i < E) {
        int c = col[i];
        atomicAdd(&msg[c], e[row[i]]);
        atomicAdd(&deg[c], 1.0f);
    }
}

// combine 1: finalize dinv (in place over deg), apply layer-1 update,
// and reset msg for pass 2 (saves a separate zero pass).
__global__ void k_combine1(float* __restrict__ e, float* __restrict__ msg,
                           float* __restrict__ deg, int n) {
    int i = blockIdx.x * blockDim.x + threadIdx.x;
    if (i < n) {
        float d  = deg[i];
        float di = (d > 0.0f) ? (1.0f / d) : 0.0f;   // nan_to_num(1/deg)
        deg[i] = di;                                  // becomes dinv for pass 2
        e[i]   = ALPHA * e[i] + (1.0f - ALPHA) * di * msg[i];
        msg[i] = 0.0f;
    }
}

// ------------------------------------------------ propagation, pass 2
__global__ void k_prop2(const int* __restrict__ row, const int* __restrict__ col,
                        const float* __restrict__ e, float* __restrict__ msg, int E) {
    int i = blockIdx.x * blockDim.x + threadIdx.x;
    if (i < E) atomicAdd(&msg[col[i]], e[row[i]]);
}

__global__ void k_combine2(float* __restrict__ e, const float* __restrict__ msg,
                           const float* __restrict__ dinv, int n) {
    int i = blockIdx.x * blockDim.x + threadIdx.x;
    if (i < n) e[i] = ALPHA * e[i] + (1.0f - ALPHA) * dinv[i] * msg[i];
}

// ------------------------------------------------ scoring: gather + WMMA MLP
// Each wave32 scores 16 gathered nodes.
//   l_occ: 2 lanes per node, 32 dims each, combined with ds_swizzle xor16.
//   Layer 1: V_WMMA_F32_16X16X4_F32  (A = feats 16x4, B = W1 padded 4x16,
//            C = b1 broadcast) -> D = 16x16 hidden pre-activation.
//   Layer 2: relu(D)*W2[lane], 16-lane ds_swizzle butterfly per D register.
__global__ void k_score(const float* __restrict__ emb, const float* __restrict__ center,
                        const float* __restrict__ W1, const float* __restrict__ b1,
                        const float* __restrict__ W2, const float* __restrict__ b2,
                        const float* __restrict__ lnll, const float* __restrict__ ener,
                        const int* __restrict__ node_idx, float* __restrict__ out,
                        int nidx) {
    int tid  = threadIdx.x;
    int lane = tid & 31;
    int s    = lane & 15;           // node slot within the wave's 16 nodes
    int half = lane >> 4;           // 0: dims 0-31 / K=0,1   1: dims 32-63 / K=2,3
    int wave = blockIdx.x * (blockDim.x >> 5) + (tid >> 5);
    int i0   = wave * 16;

    int gi  = i0 + s;
    int idx = (gi < nidx) ? node_idx[gi] : 0;   // safe dummy row when OOB

    // ---- l_occ partial over this lane's 32 dims
    const float4* ep = (const float4*)(emb + (size_t)idx * H_DIM + half * 32);
    const float4* cp = (const float4*)(center + half * 32);
    float acc = 0.0f;
#pragma unroll
    for (int j = 0; j < 8; ++j) {
        float4 a = ep[j], c = cp[j];
        float dx = a.x - c.x, dy = a.y - c.y, dz = a.z - c.z, dw = a.w - c.w;
        acc += dx * dx + dy * dy + dz * dz + dw * dw;
    }
    float occ = acc + SWZ_XOR(acc, 16);   // full 64-dim sq-distance, both halves

    // ---- A (feats, 16x4 K-padded) and B (W1 padded to 4x16), ISA 7.12.2 layout
    v2f A, B;
    if (half == 0) {
        A[0] = lnll[idx];        // K=0
        A[1] = ener[idx];        // K=1
        B[0] = W1[0 * 16 + s];   // row K=0
        B[1] = W1[1 * 16 + s];   // row K=1
    } else {
        A[0] = occ;              // K=2
        A[1] = 0.0f;             // K=3 pad
        B[0] = W1[2 * 16 + s];   // row K=2
        B[1] = 0.0f;             // row K=3 pad
    }

    float bias = b1[s];
    v8f Cm;
#pragma unroll
    for (int j = 0; j < 8; ++j) Cm[j] = bias;    // C[m][n] = b1[n]

    v8f D = __builtin_amdgcn_wmma_f32_16x16x4_f32(
        /*neg_a=*/false, A, /*neg_b=*/false, B,
        /*c_mod=*/(short)0, Cm, /*reuse_a=*/false, /*reuse_b=*/false);

    // ---- layer 2 + sigmoid; D vgpr j holds row m=j+8*half, n=s
    float w2  = W2[s];
    float b2v = b2[0];
#pragma unroll
    for (int j = 0; j < 8; ++j) {
        float t = fmaxf(D[j], 0.0f) * w2;
        t += SWZ_XOR(t, 1);
        t += SWZ_XOR(t, 2);
        t += SWZ_XOR(t, 4);
        t += SWZ_XOR(t, 8);                      // sum over n within each half
        // sigmoid via raw v_rcp_f32 (avoids the IEEE div_scale/div_fixup chain)
        float sc = __builtin_amdgcn_rcpf(1.0f + __expf(-(t + b2v)));
        int m  = j + (half << 3);
        int go = i0 + m;
        if (s == 0 && go < nidx) out[go] = sc;
    }
}

// ---------------------------------------------------------------- launcher
extern "C" void kernel_launch(void* const* d_in, const int* in_sizes, int n_in,
                              void* d_out, int out_size, void* d_ws, size_t ws_size,
                              hipStream_t stream) {
    const float* logits = (const float*)d_in[0];
    const float* emb    = (const float*)d_in[1];
    const float* center = (const float*)d_in[2];
    const float* W1     = (const float*)d_in[3];
    const float* b1     = (const float*)d_in[4];
    const float* W2     = (const float*)d_in[5];
    const float* b2     = (const float*)d_in[6];
    const int*   edges  = (const int*)d_in[7];   // [2,E], row then col
    const int*   nodeix = (const int*)d_in[8];
    float*       out    = (float*)d_out;

    const int N    = in_sizes[0] / C_CLS;
    const int E    = in_sizes[7] / 2;
    const int NIDX = in_sizes[8];

    float* w    = (float*)d_ws;     // 4*N floats = 8 MB scratch
    float* lnll = w;
    float* ener = w + (size_t)N;
    float* dinv = w + 2 * (size_t)N;   // holds deg during pass 1
    float* msg  = w + 3 * (size_t)N;   // adjacent to dinv -> one zero launch

    const int B = 256;
    dim3 bn((N + B - 1) / B), be((E + B - 1) / B), bn2((2 * N + B - 1) / B);

    k_logits  <<<bn,  B, 0, stream>>>(logits, lnll, ener, N);
    k_zero    <<<bn2, B, 0, stream>>>(dinv, 2 * N);              // deg + msg
    k_prop1   <<<be,  B, 0, stream>>>(edges, edges + E, ener, msg, dinv, E);
    k_combine1<<<bn,  B, 0, stream>>>(ener, msg, dinv, N);       // dinv finalized, msg reset
    k_prop2   <<<be,  B, 0, stream>>>(edges, edges + E, ener, msg, E);
    k_combine2<<<bn,  B, 0, stream>>>(ener, msg, dinv, N);

    const int nodes_per_block = (B / 32) * 16;   // 8 waves x 16 nodes
    dim3 bs((NIDX + nodes_per_block - 1) / nodes_per_block);
    k_score<<<bs, B, 0, stream>>>(emb, center, W1, b1, W2, b2,
                                  lnll, ener, nodeix, out, NIDX);
}